// Model_52338471469141
// MI455X (gfx1250) — compile-verified
//
#include <hip/hip_runtime.h>
#include <math.h>

typedef float v2f __attribute__((ext_vector_type(2)));
typedef float v8f __attribute__((ext_vector_type(8)));

#define N_NODES 262144
#define DIM     128
#define N_EDGES 131072
#define OUTD    128
#define LDS_PITCH 132   // 128 + 4 pad: rows 16B-aligned, conflict-free b64 fragment loads
#define STRIPS_PER_BLOCK 8

// ---------------------------------------------------------------------------
// Kernel 1: feats(ws) = raw ; acc(d_out) = raw      (grid-stride float4 copy)
// ---------------------------------------------------------------------------
__global__ void init_copy(const float4* __restrict__ raw,
                          float4* __restrict__ feats,
                          float4* __restrict__ acc, int n4) {
  int i = blockIdx.x * blockDim.x + threadIdx.x;
  int stride = gridDim.x * blockDim.x;
  for (; i < n4; i += stride) {
    float4 v = raw[i];
    feats[i] = v;
    acc[i]   = v;
  }
}

// ---------------------------------------------------------------------------
// Kernel 2: sequential edge scan. ONE wave32; lane owns float4 (4*32 = 128=D).
// Serial chain is inherent (edge i reads feats state written by earlier edges),
// so minimize per-iteration latency: no barriers (single-wave store->load to
// same address is kept in order), shfl_xor butterfly for the L2 norm, b128
// vector accesses, and global_prefetch_b8 of the next source row.
// ---------------------------------------------------------------------------
__global__ void edge_scan(const int2* __restrict__ edges,
                          const int* __restrict__ times,
                          float4* feats, float4* acc) {
  const int tid = threadIdx.x;                       // 0..31
  const float inv_all = 1.0f / (float)(times[N_EDGES - 1] + 1);

  int2 e_next = edges[0];
  int  tm_next = times[0];

  for (int e = 0; e < N_EDGES; ++e) {
    const int s  = e_next.x;
    const int t  = e_next.y;
    const int tm = tm_next;
    if (e + 1 < N_EDGES) {
      e_next  = edges[e + 1];
      tm_next = times[e + 1];
      __builtin_prefetch(&feats[(size_t)e_next.x * 32 + tid], 0, 0);
    }

    // gather source row, L2-normalize across the wave
    float4 f = feats[(size_t)s * 32 + tid];
    float ss = f.x * f.x + f.y * f.y + f.z * f.z + f.w * f.w;
    ss += __shfl_xor(ss, 1);
    ss += __shfl_xor(ss, 2);
    ss += __shfl_xor(ss, 4);
    ss += __shfl_xor(ss, 8);
    ss += __shfl_xor(ss, 16);
    const float r = 1.0f / fmaxf(sqrtf(ss), 1e-12f);

    float4 v;
    v.x = f.x * r; v.y = f.y * r; v.z = f.z * r; v.w = f.w * r;

    // feats[t] += v   (read-before-write elementwise: s==t case is correct)
    float4 ft = feats[(size_t)t * 32 + tid];
    ft.x += v.x; ft.y += v.y; ft.z += v.z; ft.w += v.w;
    feats[(size_t)t * 32 + tid] = ft;

    // acc[t] += v with element 127 (lane 31 .w) replaced by (tm+1)/all_time
    float4 v2 = v;
    if (tid == 31) v2.w = (float)(tm + 1) * inv_all;
    float4 at = acc[(size_t)t * 32 + tid];
    at.x += v2.x; at.y += v2.y; at.z += v2.z; at.w += v2.w;
    acc[(size_t)t * 32 + tid] = at;
  }
}

// ---------------------------------------------------------------------------
// Kernel 3: out = tanh(acc @ W^T), fp32 WMMA 16x16x4, in place on d_out.
// Block = 256 thr = 8 waves; block owns 8 M-strips of 16 rows (amortizes the
// one-time W staging 8x -> W global traffic 1 GB -> 128 MB). Wave w owns
// N-tile w and keeps its 32 B-fragments (v2f each) in registers for all
// strips, so the inner WMMA loop only pulls A fragments from LDS.
// A strip is double-buffered in LDS: global load of strip s+1 (into regs)
// overlaps WMMA of strip s; one barrier per strip.
// In-place safe: strip rows are read (prologue / prior iteration) strictly
// before they are written (same iteration epilogue), and blocks own disjoint
// row ranges.
// ---------------------------------------------------------------------------
__global__ void __launch_bounds__(256) gemm_tanh(float* data,
                                                 const float* __restrict__ weight) {
  __shared__ float lds_w[OUTD * LDS_PITCH];       // 67584 B
  __shared__ float lds_a[2][16 * LDS_PITCH];      // 2 x 8448 B

  const int tid = threadIdx.x;
  const size_t mbase = (size_t)blockIdx.x * (16 * STRIPS_PER_BLOCK);

  // stage W: 128x128 f32, coalesced float4, padded pitch (once per block)
  for (int i = tid; i < OUTD * DIM / 4; i += 256) {
    const int row = i >> 5, c4 = i & 31;
    float4 w = ((const float4*)weight)[(size_t)row * 32 + c4];
    *(float4*)&lds_w[row * LDS_PITCH + c4 * 4] = w;
  }
  __syncthreads();

  const int lane = tid & 31;
  const int wv   = tid >> 5;        // N-tile index 0..7
  const int l16  = lane & 15;
  const int h    = lane >> 4;

  // hoist B fragments to registers: lane l holds W^T[K=4kk+2h+{0,1}][n]
  v2f bf[32];
  {
    const float* bp = &lds_w[(wv * 16 + l16) * LDS_PITCH + 2 * h];
#pragma unroll
    for (int kk = 0; kk < 32; ++kk) bf[kk] = *(const v2f*)(bp + kk * 4);
  }

  // per-thread A staging registers: 16x128 strip / 256 threads = 2 float4
  const int row0 = tid >> 5,          c40 = tid & 31;          // idx = tid
  const int row1 = (tid + 256) >> 5,  c41 = tid & 31;          // idx = tid+256
  float4 r0 = ((const float4*)data)[(mbase + row0) * 32 + c40];
  float4 r1 = ((const float4*)data)[(mbase + row1) * 32 + c41];

  for (int s = 0; s < STRIPS_PER_BLOCK; ++s) {
    const int buf = s & 1;
    *(float4*)&lds_a[buf][row0 * LDS_PITCH + c40 * 4] = r0;
    *(float4*)&lds_a[buf][row1 * LDS_PITCH + c41 * 4] = r1;
    __syncthreads();

    if (s + 1 < STRIPS_PER_BLOCK) {   // overlap next strip's global load
      const size_t mb = mbase + (size_t)(s + 1) * 16;
      r0 = ((const float4*)data)[(mb + row0) * 32 + c40];
      r1 = ((const float4*)data)[(mb + row1) * 32 + c41];
    }

    v8f c = {};
    const float* ap = &lds_a[buf][l16 * LDS_PITCH + 2 * h];
#pragma unroll
    for (int kk = 0; kk < 32; ++kk) {
      v2f a = *(const v2f*)(ap + kk * 4);
      c = __builtin_amdgcn_wmma_f32_16x16x4_f32(false, a, false, bf[kk],
                                                (short)0, c, false, false);
    }

    // C layout: VGPR r -> rows {r, r+8}; lanes 0-15 / 16-31 are row halves
#pragma unroll
    for (int r = 0; r < 8; ++r) {
      const size_t m = mbase + (size_t)s * 16 + r + 8 * h;
      data[m * OUTD + wv * 16 + l16] = tanhf(c[r]);
    }
  }
}

// ---------------------------------------------------------------------------
extern "C" void kernel_launch(void* const* d_in, const int* in_sizes, int n_in,
                              void* d_out, int out_size, void* d_ws, size_t ws_size,
                              hipStream_t stream) {
  const float* raw    = (const float*)d_in[0];   // [N, D] f32
  const int*   edges  = (const int*)d_in[1];     // [E, 2] i32
  const int*   times  = (const int*)d_in[2];     // [E]    i32 (sorted)
  const float* weight = (const float*)d_in[3];   // [OUT, D] f32
  float* out   = (float*)d_out;                  // [N, OUT] f32; doubles as acc
  float* feats = (float*)d_ws;                   // needs N*D*4 = 128 MiB scratch

  init_copy<<<2048, 256, 0, stream>>>((const float4*)raw, (float4*)feats,
                                      (float4*)out, N_NODES * DIM / 4);
  edge_scan<<<1, 32, 0, stream>>>((const int2*)edges, times,
                                  (float4*)feats, (float4*)out);
  gemm_tanh<<<N_NODES / (16 * STRIPS_PER_BLOCK), 256, 0, stream>>>(out, weight);
}